// GAT_62843961475709
// MI455X (gfx1250) — compile-verified
//
#include <hip/hip_runtime.h>

typedef __attribute__((ext_vector_type(2))) float v2f;
typedef __attribute__((ext_vector_type(8))) float v8f;

#define NEG_SLOPE 0.2f

// ---- monotonic float <-> uint encoding for atomicMax-based segment max ----
__device__ __forceinline__ unsigned f2ord(float f) {
  unsigned b = __float_as_uint(f);
  return (b & 0x80000000u) ? ~b : (b | 0x80000000u);
}
__device__ __forceinline__ float ord2f(unsigned u) {
  unsigned b = (u & 0x80000000u) ? (u & 0x7fffffffu) : ~u;
  return __uint_as_float(b);
}

// =====================================================================
// fp32 GEMM via V_WMMA_F32_16X16X4_F32. One wave -> one 16x16 C tile.
// A[M,K] row-major, B[K,Ncol] row-major, C[M,Ncol]. M%16==0, K%4==0,
// Ncol%16==0 (true for this problem), so EXEC stays all-ones.
// A layout (16x4 f32): lanes 0-15 hold M=lane rows, VGPR0/1 = K {0,1};
// lanes 16-31 same rows, K {2,3}.  B is the mirror (N striped on lanes).
// C/D: VGPR r -> row (half*8 + r), col = lane&15.
// =====================================================================
__global__ void gat_gemm_wmma(const float* __restrict__ A,
                              const float* __restrict__ B,
                              float* __restrict__ C,
                              int K, int Ncol, int tilesN) {
  int tile_n = blockIdx.x * blockDim.y + threadIdx.y;   // wave-uniform guard
  if (tile_n >= tilesN) return;
  int tile_m = blockIdx.y;
  int lane = threadIdx.x & 31;
  int half = lane >> 4;
  int idx  = lane & 15;

  const float* arow = A + (size_t)(tile_m * 16 + idx) * (size_t)K;
  int col = tile_n * 16 + idx;

  v8f acc = {};
  for (int k0 = 0; k0 < K; k0 += 4) {
    int ka = k0 + half * 2;
    v2f a = *(const v2f*)(arow + ka);        // 8B-aligned pair load
    v2f b;
    b.x = B[(size_t)ka * Ncol + col];
    b.y = B[(size_t)(ka + 1) * Ncol + col];
    acc = __builtin_amdgcn_wmma_f32_16x16x4_f32(
        /*neg_a=*/false, a, /*neg_b=*/false, b,
        /*c_mod=*/(short)0, acc, /*reuse_a=*/false, /*reuse_b=*/false);
  }
  size_t crow = (size_t)(tile_m * 16 + half * 8);
#pragma unroll
  for (int r = 0; r < 8; ++r)
    C[(crow + r) * (size_t)Ncol + col] = acc[r];
}

// =====================================================================
// consts[0..3]=K1[h], consts[4]=K2, consts[5]=mean(edge_attr)
// =====================================================================
__global__ void gat_prep_consts(const float* __restrict__ eattr, int E,
                                const float* __restrict__ l1e,
                                const float* __restrict__ a1e,
                                const float* __restrict__ l2e,
                                const float* __restrict__ a2e,
                                float* __restrict__ consts) {
  __shared__ float red[256];
  float s = 0.0f;
  for (int i = threadIdx.x; i < E; i += 256) s += eattr[i];
  red[threadIdx.x] = s;
  __syncthreads();
  for (int off = 128; off > 0; off >>= 1) {
    if (threadIdx.x < off) red[threadIdx.x] += red[threadIdx.x + off];
    __syncthreads();
  }
  if (threadIdx.x < 4) {
    float k = 0.0f;
    for (int c = 0; c < 32; ++c)
      k += l1e[threadIdx.x * 32 + c] * a1e[threadIdx.x * 32 + c];
    consts[threadIdx.x] = k;
  }
  if (threadIdx.x == 4) {
    float k = 0.0f;
    for (int c = 0; c < 32; ++c) k += l2e[c] * a2e[c];
    consts[4] = k;
  }
  if (threadIdx.x == 5) consts[5] = red[0] / (float)E;
}

// s_src[n,h] = <xl[n,h,:], att_src[h,:]> ; s_dst likewise
__global__ void gat_node_scores(const float* __restrict__ xl,
                                const float* __restrict__ att_src,
                                const float* __restrict__ att_dst,
                                float* __restrict__ ssrc,
                                float* __restrict__ sdst,
                                int N, int H, int Cch) {
  int i = blockIdx.x * blockDim.x + threadIdx.x;
  if (i >= N * H) return;
  int n = i / H, h = i - n * H;
  const float* xp = xl + (size_t)n * H * Cch + (size_t)h * Cch;
  const float* as = att_src + h * Cch;
  const float* ad = att_dst + h * Cch;
  float s1 = 0.0f, s2 = 0.0f;
  for (int c = 0; c < Cch; ++c) {
    float v = xp[c];
    s1 += v * as[c];
    s2 += v * ad[c];
  }
  ssrc[i] = s1;
  sdst[i] = s2;
}

// alpha = leaky(s_src[src]+s_dst[dst]+ea*K[h]); segment max via atomicMax
__global__ void gat_edge_alpha(const int* __restrict__ eidx,
                               const float* __restrict__ eattr,
                               const float* __restrict__ consts, int koff,
                               const float* __restrict__ ssrc,
                               const float* __restrict__ sdst,
                               float* __restrict__ alpha,
                               unsigned* __restrict__ amax,
                               int E, int N, int H) {
  int i = blockIdx.x * blockDim.x + threadIdx.x;
  int total = (E + N) * H;
  if (i >= total) return;
  int e = i / H, h = i - e * H;
  int s, d; float ea;
  if (e < E) { s = eidx[e]; d = eidx[E + e]; ea = eattr[e]; }
  else       { s = d = e - E;                ea = consts[5]; }
  float a = ssrc[s * H + h] + sdst[d * H + h] + ea * consts[koff + h];
  a = a > 0.0f ? a : NEG_SLOPE * a;
  alpha[i] = a;
  atomicMax(&amax[d * H + h], f2ord(a));
}

// ex = exp(alpha - amax[dst]); den += ex (in-place over alpha buffer)
__global__ void gat_edge_exp(const int* __restrict__ eidx,
                             float* __restrict__ alpha,
                             const unsigned* __restrict__ amax,
                             float* __restrict__ den,
                             int E, int N, int H) {
  int i = blockIdx.x * blockDim.x + threadIdx.x;
  int total = (E + N) * H;
  if (i >= total) return;
  int e = i / H, h = i - e * H;
  int d = (e < E) ? eidx[E + e] : (e - E);
  float ex = __expf(alpha[i] - ord2f(amax[d * H + h]));
  alpha[i] = ex;
  atomicAdd(&den[d * H + h], ex);
}

// out[dst, k] += (ex/den[dst,h]) * feat[src, k]
__global__ void gat_edge_scatter(const int* __restrict__ eidx,
                                 const float* __restrict__ feat,
                                 const float* __restrict__ exbuf,
                                 const float* __restrict__ den,
                                 float* __restrict__ out,
                                 int E, int N, int H, int Cch) {
  int C = H * Cch;
  int total = (E + N) * C;
  int i = blockIdx.x * blockDim.x + threadIdx.x;
  if (i >= total) return;
  int e = i / C, k = i - e * C;
  int h = k / Cch;
  int s, d;
  if (e < E) { s = eidx[e]; d = eidx[E + e]; }
  else       { s = d = e - E; }
  float w = exbuf[e * H + h] / den[d * H + h];
  atomicAdd(&out[(size_t)d * C + k], w * feat[(size_t)s * C + k]);
}

__global__ void gat_elu_bias(float* __restrict__ buf,
                             const float* __restrict__ bias,
                             int total, int C) {
  int i = blockIdx.x * blockDim.x + threadIdx.x;
  if (i >= total) return;
  float v = buf[i] + bias[i % C];
  buf[i] = v > 0.0f ? v : (__expf(v) - 1.0f);
}

__global__ void gat_add_bias(float* __restrict__ buf,
                             const float* __restrict__ bias,
                             int total, int C) {
  int i = blockIdx.x * blockDim.x + threadIdx.x;
  if (i >= total) return;
  buf[i] += bias[i % C];
}

extern "C" void kernel_launch(void* const* d_in, const int* in_sizes, int n_in,
                              void* d_out, int out_size, void* d_ws, size_t ws_size,
                              hipStream_t stream) {
  (void)n_in; (void)ws_size;
  const float* x        = (const float*)d_in[0];
  const int*   eidx     = (const int*)d_in[1];
  const float* eattr    = (const float*)d_in[2];
  const float* lin1_w   = (const float*)d_in[3];
  const float* att1_src = (const float*)d_in[4];
  const float* att1_dst = (const float*)d_in[5];
  const float* lin1_ew  = (const float*)d_in[6];
  const float* att1_e   = (const float*)d_in[7];
  const float* bias1    = (const float*)d_in[8];
  const float* lin2_w   = (const float*)d_in[9];
  const float* att2_src = (const float*)d_in[10];
  const float* att2_dst = (const float*)d_in[11];
  const float* lin2_ew  = (const float*)d_in[12];
  const float* att2_e   = (const float*)d_in[13];
  const float* bias2    = (const float*)d_in[14];
  float* out = (float*)d_out;

  const int IN = 128, HID = 32, HEADS = 4, OUT = 32;
  const int C1 = HEADS * HID;                 // 128
  const int N = in_sizes[0] / IN;             // 50000
  const int E = in_sizes[1] / 2;              // 800000
  const int Etot = E + N;                     // 850000

  // ---- workspace carve-out (floats) ----
  float* ws = (float*)d_ws;
  size_t off = 0;
  float*    xl    = ws + off; off += (size_t)N * C1;      // layer-1 features
  float*    hbuf  = ws + off; off += (size_t)N * C1;      // layer-1 out / layer-2 in
  float*    hl    = ws + off; off += (size_t)N * OUT;     // layer-2 features
  float*    ssrc1 = ws + off; off += (size_t)N * HEADS;
  float*    sdst1 = ws + off; off += (size_t)N * HEADS;
  unsigned* amax1 = (unsigned*)(ws + off); off += (size_t)N * HEADS;
  float*    den1  = ws + off; off += (size_t)N * HEADS;
  float*    ssrc2 = ws + off; off += (size_t)N;
  float*    sdst2 = ws + off; off += (size_t)N;
  unsigned* amax2 = (unsigned*)(ws + off); off += (size_t)N;
  float*    den2  = ws + off; off += (size_t)N;
  float*    exbuf = ws + off; off += (size_t)Etot * HEADS;
  float*    consts= ws + off; off += 8;

  // ---- zero accumulators (graph-capture-safe) ----
  hipMemsetAsync(hbuf,  0, (size_t)N * C1 * sizeof(float), stream);
  hipMemsetAsync(amax1, 0, (size_t)N * HEADS * sizeof(unsigned), stream);
  hipMemsetAsync(den1,  0, (size_t)N * HEADS * sizeof(float), stream);
  hipMemsetAsync(amax2, 0, (size_t)N * sizeof(unsigned), stream);
  hipMemsetAsync(den2,  0, (size_t)N * sizeof(float), stream);
  hipMemsetAsync(out,   0, (size_t)out_size * sizeof(float), stream);

  gat_prep_consts<<<1, 256, 0, stream>>>(eattr, E, lin1_ew, att1_e, lin2_ew, att2_e, consts);

  // ---------- Layer 1 ----------
  {
    dim3 blk(32, 4);
    dim3 grd((C1 / 16 + 3) / 4, N / 16);
    gat_gemm_wmma<<<grd, blk, 0, stream>>>(x, lin1_w, xl, IN, C1, C1 / 16);
  }
  {
    int t = N * HEADS;
    gat_node_scores<<<(t + 255) / 256, 256, 0, stream>>>(xl, att1_src, att1_dst,
                                                         ssrc1, sdst1, N, HEADS, HID);
  }
  {
    int t = Etot * HEADS;
    gat_edge_alpha<<<(t + 255) / 256, 256, 0, stream>>>(eidx, eattr, consts, 0,
                                                        ssrc1, sdst1, exbuf, amax1,
                                                        E, N, HEADS);
    gat_edge_exp<<<(t + 255) / 256, 256, 0, stream>>>(eidx, exbuf, amax1, den1,
                                                      E, N, HEADS);
  }
  {
    long long tl = (long long)Etot * C1;   // ~108.8M
    gat_edge_scatter<<<(unsigned)((tl + 255) / 256), 256, 0, stream>>>(
        eidx, xl, exbuf, den1, hbuf, E, N, HEADS, HID);
  }
  {
    int t = N * C1;
    gat_elu_bias<<<(t + 255) / 256, 256, 0, stream>>>(hbuf, bias1, t, C1);
  }

  // ---------- Layer 2 ----------
  {
    dim3 blk(32, 4);
    dim3 grd((OUT / 16 + 3) / 4, N / 16);
    gat_gemm_wmma<<<grd, blk, 0, stream>>>(hbuf, lin2_w, hl, C1, OUT, OUT / 16);
  }
  {
    int t = N;
    gat_node_scores<<<(t + 255) / 256, 256, 0, stream>>>(hl, att2_src, att2_dst,
                                                         ssrc2, sdst2, N, 1, OUT);
  }
  {
    int t = Etot;
    gat_edge_alpha<<<(t + 255) / 256, 256, 0, stream>>>(eidx, eattr, consts, 4,
                                                        ssrc2, sdst2, exbuf, amax2,
                                                        E, N, 1);
    gat_edge_exp<<<(t + 255) / 256, 256, 0, stream>>>(eidx, exbuf, amax2, den2,
                                                      E, N, 1);
  }
  {
    long long tl = (long long)Etot * OUT;  // 27.2M
    gat_edge_scatter<<<(unsigned)((tl + 255) / 256), 256, 0, stream>>>(
        eidx, hl, exbuf, den2, out, E, N, 1, OUT);
  }
  {
    int t = N * OUT;
    gat_add_bias<<<(t + 255) / 256, 256, 0, stream>>>(out, bias2, t, OUT);
  }
}